// MessagePassingModel_76124000354610
// MI455X (gfx1250) — compile-verified
//
#include <hip/hip_runtime.h>
#include <math.h>

// ---------------- model constants ----------------
#define F        32
#define KB       8
#define NP       15
#define NNODE    2048
#define NEDGE    32768
#define BATCH    16
#define NPB      128          // nodes per batch
#define NITER    3
#define XSTRIDE  576          // 2*9*32 floats per node
#define CGTOT    615

typedef __attribute__((ext_vector_type(2))) float v2f;
typedef __attribute__((ext_vector_type(8))) float v8f;

// path tables: (l1,l2,l3) for LMAX=2 with triangle rule, python enumeration order
__constant__ int c_l1[NP]      = {0,0,0,1,1,1,1,1,1,2,2,2,2,2,2};
__constant__ int c_l2[NP]      = {0,1,2,0,1,1,1,2,2,0,1,1,2,2,2};
__constant__ int c_l3[NP]      = {0,1,2,1,0,1,2,1,2,2,1,2,0,1,2};
__constant__ int c_cgoff[NP+1] = {0,1,10,35,44,53,80,125,170,245,270,315,390,415,490,615};
__constant__ int c_lofm[9]     = {0,1,1,1,2,2,2,2,2};

// ---------------- CG tensor construction (device, f64, same math as reference) ----
__device__ double dfact(int k){ double f=1.0; for(int i=2;i<=k;++i) f*=(double)i; return f; }

__device__ double cg_coef(int j1,int m1,int j2,int m2,int j3,int m3){
  if (m1+m2!=m3) return 0.0;
  int lo = (j1>j2)?(j1-j2):(j2-j1);
  if (j3 < lo || j3 > j1+j2) return 0.0;
  double pre = sqrt((2.0*j3+1.0)*dfact(j1+j2-j3)*dfact(j1-j2+j3)*dfact(-j1+j2+j3)/dfact(j1+j2+j3+1));
  pre *= sqrt(dfact(j1+m1)*dfact(j1-m1)*dfact(j2+m2)*dfact(j2-m2)*dfact(j3+m3)*dfact(j3-m3));
  double s=0.0;
  for(int k=0;k<=j1+j2-j3;++k){
    int d1=j1+j2-j3-k, d2=j1-m1-k, d3=j2+m2-k, d4=j3-j2+m1+k, d5=j3-j1-m2+k;
    if (d1<0||d2<0||d3<0||d4<0||d5<0) continue;
    double den = dfact(k)*dfact(d1)*dfact(d2)*dfact(d3)*dfact(d4)*dfact(d5);
    s += ((k&1)? -1.0:1.0)/den;
  }
  return pre*s;
}

__device__ void buildV(int l, double Vr[5][5], double Vi[5][5]){
  for(int i=0;i<5;++i) for(int j=0;j<5;++j){ Vr[i][j]=0.0; Vi[i][j]=0.0; }
  const double is2 = 0.7071067811865475244;
  Vr[l][l] = 1.0;
  for(int m=1;m<=l;++m){
    double sg = (m&1)? -1.0 : 1.0;
    Vr[l-m][l+m] =  is2;
    Vi[l-m][l-m] = -is2;
    Vr[l+m][l+m] =  sg*is2;
    Vi[l+m][l-m] =  sg*is2;
  }
}

__global__ void cg_init_kernel(float* __restrict__ cgbuf){
  for (int idx = threadIdx.x; idx < CGTOT; idx += blockDim.x){
    int pi=0; while(pi<NP-1 && idx>=c_cgoff[pi+1]) ++pi;
    int l1=c_l1[pi], l2=c_l2[pi], l3=c_l3[pi];
    int n2=2*l2+1, n3=2*l3+1;
    int local = idx - c_cgoff[pi];
    int a = local/(n2*n3); int r = local%(n2*n3); int b = r/n3; int c = r%n3;
    double V1r[5][5],V1i[5][5],V2r[5][5],V2i[5][5],V3r[5][5],V3i[5][5];
    buildV(l1,V1r,V1i); buildV(l2,V2r,V2i); buildV(l3,V3r,V3i);
    double tr=0.0, ti=0.0;
    for(int m=0;m<2*l1+1;++m){
      double a1r=V1r[m][a], a1i=V1i[m][a];
      if (a1r==0.0 && a1i==0.0) continue;
      for(int nn=0;nn<n2;++nn){
        double a2r=V2r[nn][b], a2i=V2i[nn][b];
        if (a2r==0.0 && a2i==0.0) continue;
        double v12r = a1r*a2r - a1i*a2i;
        double v12i = a1r*a2i + a1i*a2r;
        for(int p=0;p<n3;++p){
          double cc = cg_coef(l1, m-l1, l2, nn-l2, l3, p-l3);
          if (cc==0.0) continue;
          double a3r=V3r[p][c], a3i=V3i[p][c];
          double zr = v12r*a3r + v12i*a3i;   // * conj(V3)
          double zi = v12i*a3r - v12r*a3i;
          tr += cc*zr; ti += cc*zi;
        }
      }
    }
    cgbuf[idx] = (float)(tr+ti);
  }
}

// ---------------- edge geometry: r, sph harmonics, radial basis ----------------
__global__ void edge_geom_kernel(const float* __restrict__ pos,
                                 const int* __restrict__ dst, const int* __restrict__ src,
                                 float* __restrict__ rbuf, float* __restrict__ shbuf,
                                 float* __restrict__ radbuf){
  int e = blockIdx.x*blockDim.x + threadIdx.x;
  if (e>=NEDGE) return;
  int si = src[e], di = dst[e];
  float dx = pos[si*3+0]-pos[di*3+0];
  float dy = pos[si*3+1]-pos[di*3+1];
  float dz = pos[si*3+2]-pos[di*3+2];
  float r = sqrtf(dx*dx+dy*dy+dz*dz + 1e-24f);
  rbuf[e] = r;
  float x = dx/r, y = dy/r, z = dz/r;
  const float s3 = 1.7320508075688772f;
  float* sh = shbuf + e*9;
  sh[0]=1.f; sh[1]=y; sh[2]=z; sh[3]=x;
  sh[4]=s3*x*y; sh[5]=s3*y*z; sh[6]=0.5f*(3.f*z*z-1.f); sh[7]=s3*x*z; sh[8]=0.5f*s3*(x*x-y*y);
  float u = 1.f/(1.f+r);
  float w = 1.f-u;
  float wp[8]; wp[0]=1.f;
  for(int i=1;i<8;++i) wp[i]=wp[i-1]*w;
  float t = fminf(r*0.2f, 1.f-1e-6f);
  float cut = (r<5.f)? expf(1.f - 1.f/(1.f-t*t)) : 0.f;
  const float binom[8]={1.f,7.f,21.f,35.f,35.f,21.f,7.f,1.f};
  float* rb = radbuf + e*KB;
  float up = 1.f;
  for(int i=0;i<8;++i){ rb[i] = binom[i]*up*wp[7-i]*cut; up*=u; }
}

// ---------------- node init: x embed, xEF field features ----------------
__global__ void node_init_kernel(const float* __restrict__ embed, const int* __restrict__ Z,
                                 const float* __restrict__ Ef,
                                 float* __restrict__ x, float* __restrict__ xef){
  int tid = blockIdx.x*blockDim.x + threadIdx.x;
  if (tid>=NNODE*F) return;
  int node = tid>>5; int f = tid&31;
  float* xp = x   + node*XSTRIDE + f;
  float* ep = xef + node*XSTRIDE + f;
  int b = node>>7;
  float ef4[4] = {1.f, Ef[b*3+0]*0.001f, Ef[b*3+1]*0.001f, Ef[b*3+2]*0.001f};
  for(int p=0;p<2;++p)
    for(int m=0;m<9;++m){
      xp[(p*9+m)*F] = 0.f;
      ep[(p*9+m)*F] = (m<4)? ef4[m] : 0.f;
    }
  xp[0] = embed[Z[node]*F + f];
}

__global__ void zero_kernel(float* __restrict__ p, int count){
  int i = blockIdx.x*blockDim.x + threadIdx.x;
  if (i<count) p[i]=0.f;
}

// ---------------- message pass: one wave per edge, lane == feature ----------------
__global__ __launch_bounds__(128) void msg_kernel(const float* __restrict__ x,
    const float* __restrict__ shbuf, const float* __restrict__ radbuf,
    const int* __restrict__ dst, const int* __restrict__ src,
    const float* __restrict__ mpw, const float* __restrict__ cgbuf,
    float* __restrict__ y){
  __shared__ float sCG[CGTOT];
  for(int i=threadIdx.x;i<CGTOT;i+=blockDim.x) sCG[i]=cgbuf[i];
  __syncthreads();
  int e = blockIdx.x*4 + (threadIdx.x>>5);
  int f = threadIdx.x & 31;
  if (e>=NEDGE) return;
  float rad[KB];
  for(int k=0;k<KB;++k) rad[k]=radbuf[e*KB+k];
  float shv[9];
  for(int j=0;j<9;++j) shv[j]=shbuf[e*9+j];
  const float* xs = x + src[e]*XSTRIDE;
  float acc0[9]={0,0,0,0,0,0,0,0,0};
  float acc1[9]={0,0,0,0,0,0,0,0,0};
  for(int pi=0;pi<NP;++pi){
    int l1=c_l1[pi], l2=c_l2[pi], l3=c_l3[pi];
    int n1=2*l1+1, n2=2*l2+1, n3=2*l3+1;
    int off=c_cgoff[pi], b1=l1*l1, b2=l2*l2, b3=l3*l3;
    int flip=(l1+l2+l3)&1;
    float wr=0.f;
    const float* wp = mpw + pi*KB*F + f;   // (NP,K,F)
#pragma unroll
    for(int k=0;k<KB;++k) wr += rad[k]*wp[k*F];
    for(int a=0;a<n1;++a){
      float xa0 = xs[(b1+a)*F + f];        // parity 0
      float xa1 = xs[(9+b1+a)*F + f];      // parity 1
      for(int c=0;c<n3;++c){
        float A=0.f;
        for(int bb=0;bb<n2;++bb) A += shv[b2+bb]*sCG[off+(a*n2+bb)*n3+c];
        float t = A*wr;
        int m=b3+c;
        if(!flip){ acc0[m]+=t*xa0; acc1[m]+=t*xa1; }
        else     { acc1[m]+=t*xa0; acc0[m]+=t*xa1; }
      }
    }
  }
  float* yd = y + dst[e]*XSTRIDE + f;
  for(int m=0;m<9;++m){
    atomicAdd(yd + m*F,     acc0[m]);
    atomicAdd(yd + (9+m)*F, acc1[m]);
  }
}

// ---------------- node update: esilu(x+y), tensor(x,xEF), u = x + xEF_new -------
__global__ __launch_bounds__(256) void node_update_kernel(const float* __restrict__ x,
    const float* __restrict__ y, float* __restrict__ xef,
    const float* __restrict__ tw, const float* __restrict__ cgbuf,
    float* __restrict__ u){
  __shared__ float sCG[CGTOT];
  for(int i=threadIdx.x;i<CGTOT;i+=blockDim.x) sCG[i]=cgbuf[i];
  __syncthreads();
  int node = blockIdx.x*8 + (threadIdx.x>>5);
  int f = threadIdx.x & 31;
  if (node>=NNODE) return;
  const float* xp = x + node*XSTRIDE + f;
  const float* yp = y + node*XSTRIDE + f;
  float xn0[9], xn1[9];
  for(int m=0;m<9;++m){ xn0[m]=xp[m*F]+yp[m*F]; xn1[m]=xp[(9+m)*F]+yp[(9+m)*F]; }
  // esilu
  float s = xn0[0];
  float sig = 1.f/(1.f+expf(-s));
  float gate = sig*(1.f + s*(1.f-sig));
  for(int m=0;m<9;++m){ xn0[m]*=gate; xn1[m]*=gate; }
  xn0[0] = s*sig;
  // tensor product with xEF
  float* ep = xef + node*XSTRIDE + f;
  float xe0[9], xe1[9];
  for(int m=0;m<9;++m){ xe0[m]=ep[m*F]; xe1[m]=ep[(9+m)*F]; }
  float o0[9]={0,0,0,0,0,0,0,0,0};
  float o1[9]={0,0,0,0,0,0,0,0,0};
  for(int pi=0;pi<NP;++pi){
    int l1=c_l1[pi], l2=c_l2[pi], l3=c_l3[pi];
    int n1=2*l1+1, n2=2*l2+1, n3=2*l3+1;
    int off=c_cgoff[pi], b1=l1*l1, b2=l2*l2, b3=l3*l3;
    int sp=(l1+l2+l3)&1;
    float w00=tw[(0*NP+pi)*F+f];          // (p,q,pi,f) layout
    float w01=tw[(1*NP+pi)*F+f];
    float w10=tw[(2*NP+pi)*F+f];
    float w11=tw[(3*NP+pi)*F+f];
    for(int c=0;c<n3;++c){
      float b00=0.f,b01=0.f,b10=0.f,b11=0.f;
      for(int a=0;a<n1;++a){
        float p0=xn0[b1+a], p1=xn1[b1+a];
        for(int bb=0;bb<n2;++bb){
          float cgv = sCG[off+(a*n2+bb)*n3+c];
          float e0 = cgv*xe0[b2+bb], e1 = cgv*xe1[b2+bb];
          b00+=p0*e0; b01+=p0*e1; b10+=p1*e0; b11+=p1*e1;
        }
      }
      float even = b00*w00 + b11*w11;
      float odd  = b01*w01 + b10*w10;
      int m=b3+c;
      if(!sp){ o0[m]+=even; o1[m]+=odd; } else { o1[m]+=even; o0[m]+=odd; }
    }
  }
  float* up = u + node*XSTRIDE + f;
  for(int m=0;m<9;++m){
    ep[m*F]=o0[m]; ep[(9+m)*F]=o1[m];                  // new xEF
    up[m*F]=xn0[m]+o0[m]; up[(9+m)*F]=xn1[m]+o1[m];    // x + xEF -> degree_dense input
  }
}

// ---------------- FP32 WMMA 16x16x4 GEMM core: 16 rows x 32 cols, K=32 ----------
__device__ inline void wmma_gemm16x32(const float* __restrict__ arow,
                                      const float* __restrict__ W,
                                      int kh, v8f& c0, v8f& c1){
  int M = threadIdx.x & 15;
#pragma unroll
  for(int kk=0;kk<8;++kk){
    int k0 = kk*4 + kh*2;          // this lane's K pair within the 16x4 A tile
    v2f a; a.x = arow[k0]; a.y = arow[k0+1];
    v2f b0, b1;
    b0.x = W[k0*F + M];        b0.y = W[(k0+1)*F + M];
    b1.x = W[k0*F + 16 + M];   b1.y = W[(k0+1)*F + 16 + M];
    c0 = __builtin_amdgcn_wmma_f32_16x16x4_f32(false, a, false, b0, (short)0, c0, false, false);
    c1 = __builtin_amdgcn_wmma_f32_16x16x4_f32(false, a, false, b1, (short)0, c1, false, false);
  }
}

// Batched degree_dense for one iteration: grid = (NNODE/16, 18).
// group g = blockIdx.y -> (p = g/9, m = g%9), weight td_w_iter[(lofm[m]*2+p)].
// x[node,p,m,:] = u[node,p,m,:] @ W + y[node,p,m,:]
__global__ __launch_bounds__(32) void dd_wmma_iter_kernel(const float* __restrict__ u,
    const float* __restrict__ td_w_iter, const float* __restrict__ y,
    float* __restrict__ x){
  int g = blockIdx.y;
  int p = g/9, m = g%9;
  long off = (long)(p*9+m)*F;
  const float* W = td_w_iter + (long)(c_lofm[m]*2+p)*F*F;
  int lane = threadIdx.x;
  long row0 = (long)blockIdx.x*16;
  int M  = lane & 15;
  int kh = lane >> 4;
  v8f c0 = {0,0,0,0,0,0,0,0};
  v8f c1 = {0,0,0,0,0,0,0,0};
  const float* arow = u + off + (row0+M)*XSTRIDE;
  wmma_gemm16x32(arow, W, kh, c0, c1);
#pragma unroll
  for(int r=0;r<8;++r){
    long mr = row0 + r + kh*8;     // C layout: VGPR r -> M=r (lanes 0-15) / M=r+8 (lanes 16-31)
    const float* ap = y + off + mr*XSTRIDE;
    float* op = x + off + mr*XSTRIDE;
    op[M]    = c0[r] + ap[M];
    op[16+M] = c1[r] + ap[16+M];
  }
}

// Batched dipole degree_dense: grid = (NNODE/16, 4); p=0, m = blockIdx.y.
// xd[node,m,:] = x[node,0,m,:] @ dip_td_w[lofm[m]][0]
__global__ __launch_bounds__(32) void dd_wmma_dip_kernel(const float* __restrict__ x,
    const float* __restrict__ dip_td_w, float* __restrict__ xd){
  int m = blockIdx.y;
  const float* W = dip_td_w + (long)c_lofm[m]*F*F;
  int lane = threadIdx.x;
  long row0 = (long)blockIdx.x*16;
  int M  = lane & 15;
  int kh = lane >> 4;
  v8f c0 = {0,0,0,0,0,0,0,0};
  v8f c1 = {0,0,0,0,0,0,0,0};
  const float* arow = x + (long)m*F + (row0+M)*XSTRIDE;
  wmma_gemm16x32(arow, W, kh, c0, c1);
#pragma unroll
  for(int r=0;r<8;++r){
    long mr = row0 + r + kh*8;
    float* op = xd + (long)m*F + mr*(4*F);
    op[M]    = c0[r];
    op[16+M] = c1[r];
  }
}

// ---------------- readout helpers ----------------
__device__ inline float wave_allsum(float v){
  for(int o=16;o>0;o>>=1) v += __shfl_xor(v, o, 32);
  return v;
}

__global__ void posmean_kernel(const float* __restrict__ pos, float* __restrict__ pm){
  __shared__ float s[NPB][3];
  int b = blockIdx.x, t = threadIdx.x;
  int node = b*NPB + t;
  s[t][0]=pos[node*3]; s[t][1]=pos[node*3+1]; s[t][2]=pos[node*3+2];
  __syncthreads();
  for(int o=64;o>0;o>>=1){
    if(t<o){ s[t][0]+=s[t+o][0]; s[t][1]+=s[t+o][1]; s[t][2]+=s[t+o][2]; }
    __syncthreads();
  }
  if(t==0){
    pm[b*3+0]=s[0][0]*(1.f/NPB);
    pm[b*3+1]=s[0][1]*(1.f/NPB);
    pm[b*3+2]=s[0][2]*(1.f/NPB);
  }
}

__global__ __launch_bounds__(256) void readout_kernel(const float* __restrict__ x,
    const float* __restrict__ xd, const int* __restrict__ Z,
    const float* __restrict__ pos, const float* __restrict__ pm,
    const int* __restrict__ segs,
    const float* __restrict__ charge_w, const float* __restrict__ charge_b,
    const float* __restrict__ energy_w, const float* __restrict__ energy_b,
    const float* __restrict__ elem_bias, const float* __restrict__ ddw,
    float* __restrict__ qbuf, float* __restrict__ e_acc, float* __restrict__ d_acc){
  int node = blockIdx.x*8 + (threadIdx.x>>5);
  int f = threadIdx.x & 31;
  if (node>=NNODE) return;
  float x00 = x[node*XSTRIDE + f];                 // x[:,0,0,:]
  float qd  = wave_allsum(x00*charge_w[f]) + charge_b[0];
  float qs  = 1.f/(1.f+expf(-qd));
  float q   = qd*qs;                               // silu
  float aed = wave_allsum(x00*energy_w[f]) + energy_b[0];
  int z = Z[node];
  float ae = aed + elem_bias[z];
  // dipole feature path: xd (n,4,F) already degree_dense'd; esilu in-lane
  const float* xdp = xd + node*4*F;
  float v0=xdp[f], v1=xdp[F+f], v2=xdp[2*F+f], v3=xdp[3*F+f];
  float sg = 1.f/(1.f+expf(-v0));
  float gt = sg*(1.f + v0*(1.f-sg));
  float xd2_0 = wave_allsum(v0*sg*ddw[f]);          // LOFM[0]=0
  float xd2_1 = wave_allsum(v1*gt*ddw[F+f]);        // LOFM[1..3]=1
  float xd2_2 = wave_allsum(v2*gt*ddw[F+f]);
  float xd2_3 = wave_allsum(v3*gt*ddw[F+f]);
  float sg2 = 1.f/(1.f+expf(-xd2_0));
  float gt2 = sg2*(1.f + xd2_0*(1.f-sg2));
  int b = segs[node];
  if (f<3){
    float adip = (f==0? xd2_1 : (f==1? xd2_2 : xd2_3))*gt2;
    float cen  = pos[node*3+f] - pm[b*3+f];
    atomicAdd(&d_acc[b*3+f], q*cen + adip);
  }
  if (f==0){
    qbuf[node]=q;
    if (z>0) atomicAdd(&e_acc[b], ae);
  }
}

__global__ void coulomb_kernel(const float* __restrict__ q, const float* __restrict__ rbuf,
    const int* __restrict__ dst, const int* __restrict__ src,
    const int* __restrict__ segs, float* __restrict__ coul){
  int e = blockIdx.x*blockDim.x + threadIdx.x;
  if (e>=NEDGE) return;
  float rb = rbuf[e]*1.8897261258369282f;
  float pc = q[src[e]]*q[dst[e]]/(rb+1e-10f);
  atomicAdd(&coul[segs[dst[e]]], pc);
}

__global__ void final_kernel(const float* __restrict__ e_acc, const float* __restrict__ d_acc,
                             const float* __restrict__ coul, float* __restrict__ out){
  int t = threadIdx.x;
  if (t<BATCH)       out[t] = e_acc[t] - coul[t]*0.5f*27.211386245988f;
  else if (t<BATCH+BATCH*3) out[t] = d_acc[t-BATCH];
}

// ---------------- host launch ----------------
// workspace layout (floats)
#define OFF_CG   0L
#define OFF_R    640L
#define OFF_SH   33408L
#define OFF_RAD  328320L
#define OFF_X    590464L
#define OFF_XEF  1770112L
#define OFF_Y    2949760L
#define OFF_U    4129408L
#define OFF_XD   5309056L
#define OFF_Q    5571200L
#define OFF_PM   5573248L
#define OFF_ACC  5573296L   // eacc(16) dacc(48) coul(16)
#define WS_TOTAL 5573376L

extern "C" void kernel_launch(void* const* d_in, const int* in_sizes, int n_in,
                              void* d_out, int out_size, void* d_ws, size_t ws_size,
                              hipStream_t stream) {
  (void)in_sizes; (void)n_in; (void)out_size;
  const int*   Z        = (const int*)  d_in[0];
  const float* pos      = (const float*)d_in[1];
  const float* Ef       = (const float*)d_in[2];
  const int*   dst      = (const int*)  d_in[3];
  const int*   src      = (const int*)  d_in[4];
  const int*   segs     = (const int*)  d_in[5];
  // d_in[6] = batch_size (compile-time constant here)
  const float* embed    = (const float*)d_in[7];
  const float* mp_w     = (const float*)d_in[8];
  const float* tensor_w = (const float*)d_in[9];
  const float* td_w     = (const float*)d_in[10];
  const float* charge_w = (const float*)d_in[11];
  const float* charge_b = (const float*)d_in[12];
  const float* dip_td_w = (const float*)d_in[13];
  const float* dip_dw   = (const float*)d_in[14];
  const float* energy_w = (const float*)d_in[15];
  const float* energy_b = (const float*)d_in[16];
  const float* ebias    = (const float*)d_in[17];

  if (ws_size < (size_t)WS_TOTAL*sizeof(float)) return;
  float* ws   = (float*)d_ws;
  float* cg   = ws+OFF_CG;
  float* rb   = ws+OFF_R;
  float* sh   = ws+OFF_SH;
  float* rad  = ws+OFF_RAD;
  float* x    = ws+OFF_X;
  float* xef  = ws+OFF_XEF;
  float* y    = ws+OFF_Y;
  float* u    = ws+OFF_U;
  float* xd   = ws+OFF_XD;
  float* q    = ws+OFF_Q;
  float* pm   = ws+OFF_PM;
  float* eacc = ws+OFF_ACC;
  float* dacc = eacc+16;
  float* coul = eacc+64;

  cg_init_kernel<<<1,256,0,stream>>>(cg);
  zero_kernel<<<1,128,0,stream>>>(eacc, 80);
  edge_geom_kernel<<<NEDGE/256,256,0,stream>>>(pos,dst,src,rb,sh,rad);
  node_init_kernel<<<(NNODE*F)/256,256,0,stream>>>(embed,Z,Ef,x,xef);

  for(int i=0;i<NITER;++i){
    zero_kernel<<<(NNODE*XSTRIDE+255)/256,256,0,stream>>>(y, NNODE*XSTRIDE);
    msg_kernel<<<NEDGE/4,128,0,stream>>>(x,sh,rad,dst,src, mp_w + (long)i*NP*KB*F, cg, y);
    node_update_kernel<<<NNODE/8,256,0,stream>>>(x,y,xef, tensor_w + (long)i*4*NP*F, cg, u);
    dd_wmma_iter_kernel<<<dim3(NNODE/16,18),32,0,stream>>>(u, td_w + (long)i*3*2*F*F, y, x);
  }

  // dipole degree_dense: p=0, m=0..3, weights dip_td_w[LOFM[m]][0]
  dd_wmma_dip_kernel<<<dim3(NNODE/16,4),32,0,stream>>>(x, dip_td_w, xd);

  posmean_kernel<<<BATCH,NPB,0,stream>>>(pos,pm);
  readout_kernel<<<NNODE/8,256,0,stream>>>(x,xd,Z,pos,pm,segs,
      charge_w,charge_b,energy_w,energy_b,ebias,dip_dw, q,eacc,dacc);
  coulomb_kernel<<<NEDGE/256,256,0,stream>>>(q,rb,dst,src,segs,coul);
  final_kernel<<<1,64,0,stream>>>(eacc,dacc,coul,(float*)d_out);
}